// SelfAttention_16767552324216
// MI455X (gfx1250) — compile-verified
//
#include <hip/hip_runtime.h>
#include <hip/hip_bf16.h>

typedef __attribute__((ext_vector_type(2))) float v2f;
typedef __attribute__((ext_vector_type(8))) float v8f;

#define BATCH 64
#define CCH   256
#define HWN   1024
#define MQKV  320
#define LN_EPS 1e-5f

__device__ __forceinline__ v8f zero8() {
  v8f z = {0.f,0.f,0.f,0.f,0.f,0.f,0.f,0.f};
  return z;
}

// D = A(16x4,f32) * B(4x16,f32) + C(16x16,f32), full f32 precision
__device__ __forceinline__ v8f wmma4(v2f a, v2f b, v8f c) {
  return __builtin_amdgcn_wmma_f32_16x16x4_f32(false, a, false, b, (short)0, c, false, false);
}

// ---------------------------------------------------------------------------
// Kernel 0: zero the stats buffer
// ---------------------------------------------------------------------------
__global__ void zero_kernel(float* p, int n) {
  int i = blockIdx.x * blockDim.x + threadIdx.x;
  if (i < n) p[i] = 0.0f;
}

// ---------------------------------------------------------------------------
// Kernel 1: QKV projection.  P[b][o][n] for o in [0,320):
//   rows 0..31  : (Wq x + bq) * scale        (this is q^T, [CK,HW])
//   rows 32..63 : Wk x + bk                  ([CK,HW])
//   rows 64..319: Wv x + bv                  ([C,HW])
// Grid: (4 ntile, 5 mtile, 64 batch), 256 threads (8 waves, 2x4 wave grid),
// wave tile 32x64, workgroup tile 64x256, K chunks of 32.
// ---------------------------------------------------------------------------
__global__ __launch_bounds__(256)
void qkv_kernel(const float* __restrict__ x,
                const float* __restrict__ Wq, const float* __restrict__ bq,
                const float* __restrict__ Wk, const float* __restrict__ bk,
                const float* __restrict__ Wv, const float* __restrict__ bv,
                float* __restrict__ qkv) {
  __shared__ float As[64][33];
  __shared__ float Bs[32][257];

  const int nt   = blockIdx.x;   // 0..3
  const int mt   = blockIdx.y;   // 0..4
  const int b    = blockIdx.z;
  const int tid  = threadIdx.x;
  const int lane = tid & 31;
  const int wv   = tid >> 5;
  const int wm   = wv >> 2;      // 0..1
  const int wn   = wv & 3;       // 0..3
  const int nl   = lane & 15;
  const int half = lane >> 4;
  const float scale = 0.17677669529663687f;  // 32^-0.5

  v8f acc[2][4];
  for (int i = 0; i < 2; i++)
    for (int j = 0; j < 4; j++) acc[i][j] = zero8();

  const float* xb = x + (size_t)b * CCH * HWN;

  for (int kc = 0; kc < CCH; kc += 32) {
    __syncthreads();
    for (int e = tid; e < 64 * 32; e += 256) {
      int m = e >> 5, k = e & 31;
      int o = mt * 64 + m;
      int c = kc + k;
      float w;
      if (o < 32)      w = Wq[o * CCH + c];
      else if (o < 64) w = Wk[(o - 32) * CCH + c];
      else             w = Wv[(o - 64) * CCH + c];
      As[m][k] = w;
    }
    for (int e = tid; e < 32 * 256; e += 256) {
      int kk = e >> 8, n = e & 255;
      Bs[kk][n] = xb[(size_t)(kc + kk) * HWN + nt * 256 + n];
    }
    __syncthreads();
    for (int k4 = 0; k4 < 32; k4 += 4) {
      int ak = k4 + 2 * half;
      v2f a[2], bb[4];
      for (int im = 0; im < 2; im++) {
        a[im].x = As[wm * 32 + im * 16 + nl][ak];
        a[im].y = As[wm * 32 + im * 16 + nl][ak + 1];
      }
      for (int in = 0; in < 4; in++) {
        bb[in].x = Bs[ak][wn * 64 + in * 16 + nl];
        bb[in].y = Bs[ak + 1][wn * 64 + in * 16 + nl];
      }
      for (int im = 0; im < 2; im++)
        for (int in = 0; in < 4; in++)
          acc[im][in] = wmma4(a[im], bb[in], acc[im][in]);
    }
  }

  float* qb = qkv + (size_t)b * MQKV * HWN;
  for (int im = 0; im < 2; im++) {
    for (int in = 0; in < 4; in++) {
      int n = nt * 256 + wn * 64 + in * 16 + nl;
      for (int r = 0; r < 8; r++) {
        int o = mt * 64 + wm * 32 + im * 16 + r + 8 * half;
        float v = acc[im][in][r];
        if (o < 32)      v = (v + bq[o]) * scale;
        else if (o < 64) v = v + bk[o - 32];
        else             v = v + bv[o - 64];
        qb[(size_t)o * HWN + n] = v;
      }
    }
  }
}

// ---------------------------------------------------------------------------
// Kernel 2: attention.  One WG per (batch, 64-column tile j).
// softmax over axis i (per column).  Two sweeps over i-chunks of 64:
//   sweep A: column max.   sweep B: exp(s-max) -> LDS, column sums,
//   fused out += v_chunk @ attn_chunk.  Epilogue divides by colSum.
// ---------------------------------------------------------------------------
__global__ __launch_bounds__(256)
void attn_kernel(const float* __restrict__ qkv, float* __restrict__ attn_out) {
  __shared__ float kt[32][65];     // k[o][jlocal]
  __shared__ float qt[64][33];     // q^T chunk: [ilocal][o]
  __shared__ float sc[64][65];     // exp-scores chunk [ilocal][jlocal]
  __shared__ float vt[256][65];    // v chunk [c][ilocal]
  __shared__ float colMax[64];
  __shared__ float colSum[64];
  __shared__ float red[4][64];

  const int jt   = blockIdx.x;     // 0..15
  const int b    = blockIdx.y;
  const int tid  = threadIdx.x;
  const int lane = tid & 31;
  const int wv   = tid >> 5;
  const int nl   = lane & 15;
  const int half = lane >> 4;

  const float* qb = qkv + (size_t)b * MQKV * HWN;
  const float* kb = qb + 32 * HWN;
  const float* vb = qb + 64 * HWN;

  for (int e = tid; e < 32 * 64; e += 256) {
    int o = e >> 6, j = e & 63;
    kt[o][j] = kb[(size_t)o * HWN + jt * 64 + j];
  }
  if (tid < 64) colMax[tid] = -3.0e38f;

  const int wms = wv >> 1;         // 0..3 -> 16 score rows
  const int wns = wv & 1;          // 0..1 -> 32 score cols
  const int sRow  = wms * 16;
  const int sColB = wns * 32;

  // ---- sweep A: column max ----
  for (int ic = 0; ic < 16; ic++) {
    __syncthreads();
    for (int e = tid; e < 32 * 64; e += 256) {
      int o = e >> 6, i = e & 63;
      qt[i][o] = qb[(size_t)o * HWN + ic * 64 + i];
    }
    __syncthreads();
    v8f s[2]; s[0] = zero8(); s[1] = zero8();
    for (int k4 = 0; k4 < 32; k4 += 4) {
      int ak = k4 + 2 * half;
      v2f a; a.x = qt[sRow + nl][ak]; a.y = qt[sRow + nl][ak + 1];
      for (int in = 0; in < 2; in++) {
        v2f bb; bb.x = kt[ak][sColB + in * 16 + nl];
        bb.y = kt[ak + 1][sColB + in * 16 + nl];
        s[in] = wmma4(a, bb, s[in]);
      }
    }
    for (int in = 0; in < 2; in++) {
      float m = s[in][0];
      for (int r = 1; r < 8; r++) m = fmaxf(m, s[in][r]);
      m = fmaxf(m, __shfl_xor(m, 16));
      if (half == 0) red[wms][sColB + in * 16 + nl] = m;
    }
    __syncthreads();
    if (tid < 64) {
      float m = fmaxf(fmaxf(red[0][tid], red[1][tid]),
                      fmaxf(red[2][tid], red[3][tid]));
      colMax[tid] = fmaxf(colMax[tid], m);
    }
  }
  __syncthreads();
  if (tid < 64) colSum[tid] = 0.0f;

  // ---- sweep B: exp + sums + v @ attn ----
  const int cB = wv * 32;          // this wave's 32 channel rows of out
  v8f acc[2][4];
  for (int i = 0; i < 2; i++)
    for (int j = 0; j < 4; j++) acc[i][j] = zero8();

  for (int ic = 0; ic < 16; ic++) {
    __syncthreads();
    for (int e = tid; e < 32 * 64; e += 256) {
      int o = e >> 6, i = e & 63;
      qt[i][o] = qb[(size_t)o * HWN + ic * 64 + i];
    }
    for (int e = tid; e < 256 * 64; e += 256) {
      int c = e >> 6, i = e & 63;
      vt[c][i] = vb[(size_t)c * HWN + ic * 64 + i];
    }
    __syncthreads();
    // scores for this chunk
    v8f s[2]; s[0] = zero8(); s[1] = zero8();
    for (int k4 = 0; k4 < 32; k4 += 4) {
      int ak = k4 + 2 * half;
      v2f a; a.x = qt[sRow + nl][ak]; a.y = qt[sRow + nl][ak + 1];
      for (int in = 0; in < 2; in++) {
        v2f bb; bb.x = kt[ak][sColB + in * 16 + nl];
        bb.y = kt[ak + 1][sColB + in * 16 + nl];
        s[in] = wmma4(a, bb, s[in]);
      }
    }
    for (int in = 0; in < 2; in++) {
      int col = sColB + in * 16 + nl;
      float cm = colMax[col];
      for (int r = 0; r < 8; r++) {
        sc[sRow + r + 8 * half][col] = expf(s[in][r] - cm);
      }
    }
    __syncthreads();
    if (tid < 64) {
      float sum = 0.0f;
      for (int i = 0; i < 64; i++) sum += sc[i][tid];
      colSum[tid] += sum;
    }
    // out(32x64) += v_chunk(32x64) @ attn_chunk(64x64)
    for (int k4 = 0; k4 < 64; k4 += 4) {
      int ak = k4 + 2 * half;
      v2f a[2], bb[4];
      for (int im = 0; im < 2; im++) {
        a[im].x = vt[cB + im * 16 + nl][ak];
        a[im].y = vt[cB + im * 16 + nl][ak + 1];
      }
      for (int in = 0; in < 4; in++) {
        bb[in].x = sc[ak][in * 16 + nl];
        bb[in].y = sc[ak + 1][in * 16 + nl];
      }
      for (int im = 0; im < 2; im++)
        for (int in = 0; in < 4; in++)
          acc[im][in] = wmma4(a[im], bb[in], acc[im][in]);
    }
  }
  __syncthreads();

  float* ob = attn_out + (size_t)b * CCH * HWN + jt * 64;
  for (int im = 0; im < 2; im++) {
    for (int in = 0; in < 4; in++) {
      int n = in * 16 + nl;
      float inv = 1.0f / colSum[n];
      for (int r = 0; r < 8; r++) {
        int c = cB + im * 16 + r + 8 * half;
        ob[(size_t)c * HWN + n] = acc[im][in][r] * inv;
      }
    }
  }
}

// ---------------------------------------------------------------------------
// Kernel 3: token MLP + residual + per-batch sum/sumsq stats.
// One WG per 64-token tile of one batch.  y (pre-LN) written to d_out.
// ---------------------------------------------------------------------------
__global__ __launch_bounds__(256)
void mlp_kernel(const float* __restrict__ attn_out, const float* __restrict__ x,
                const float* __restrict__ W1, const float* __restrict__ b1,
                const float* __restrict__ W2, const float* __restrict__ b2,
                float* __restrict__ y, float* __restrict__ stats) {
  __shared__ float Ts[64][257];   // tokens [tok][c]
  __shared__ float Ws[64][257];   // weight K-chunk [k][n]
  __shared__ float Hs[64][257];   // hidden [tok][n]

  const int tt   = blockIdx.x;    // 0..15
  const int b    = blockIdx.y;
  const int tid  = threadIdx.x;
  const int lane = tid & 31;
  const int wv   = tid >> 5;
  const int nl   = lane & 15;
  const int half = lane >> 4;
  const int nB   = wv * 32;       // this wave's 32 output columns

  const float* ab = attn_out + (size_t)b * CCH * HWN + tt * 64;
  for (int e = tid; e < 256 * 64; e += 256) {
    int c = e >> 6, tok = e & 63;
    Ts[tok][c] = ab[(size_t)c * HWN + tok];
  }

  v8f acc[4][2];
  for (int i = 0; i < 4; i++)
    for (int j = 0; j < 2; j++) acc[i][j] = zero8();

  // H = relu(T @ W1 + b1)
  for (int kc = 0; kc < 256; kc += 64) {
    __syncthreads();
    for (int e = tid; e < 64 * 256; e += 256) {
      int kk = e >> 8, n = e & 255;
      Ws[kk][n] = W1[(size_t)(kc + kk) * CCH + n];
    }
    __syncthreads();
    for (int k4 = 0; k4 < 64; k4 += 4) {
      int ak = k4 + 2 * half;
      v2f a[4], bb[2];
      for (int im = 0; im < 4; im++) {
        a[im].x = Ts[im * 16 + nl][kc + ak];
        a[im].y = Ts[im * 16 + nl][kc + ak + 1];
      }
      for (int in = 0; in < 2; in++) {
        bb[in].x = Ws[ak][nB + in * 16 + nl];
        bb[in].y = Ws[ak + 1][nB + in * 16 + nl];
      }
      for (int im = 0; im < 4; im++)
        for (int in = 0; in < 2; in++)
          acc[im][in] = wmma4(a[im], bb[in], acc[im][in]);
    }
  }
  __syncthreads();
  for (int im = 0; im < 4; im++) {
    for (int in = 0; in < 2; in++) {
      int n = nB + in * 16 + nl;
      float bias = b1[n];
      for (int r = 0; r < 8; r++) {
        int m = im * 16 + r + 8 * half;
        Hs[m][n] = fmaxf(acc[im][in][r] + bias, 0.0f);
      }
    }
  }
  for (int i = 0; i < 4; i++)
    for (int j = 0; j < 2; j++) acc[i][j] = zero8();

  // Y = H @ W2 + b2 + x
  for (int kc = 0; kc < 256; kc += 64) {
    __syncthreads();
    for (int e = tid; e < 64 * 256; e += 256) {
      int kk = e >> 8, n = e & 255;
      Ws[kk][n] = W2[(size_t)(kc + kk) * CCH + n];
    }
    __syncthreads();
    for (int k4 = 0; k4 < 64; k4 += 4) {
      int ak = k4 + 2 * half;
      v2f a[4], bb[2];
      for (int im = 0; im < 4; im++) {
        a[im].x = Hs[im * 16 + nl][kc + ak];
        a[im].y = Hs[im * 16 + nl][kc + ak + 1];
      }
      for (int in = 0; in < 2; in++) {
        bb[in].x = Ws[ak][nB + in * 16 + nl];
        bb[in].y = Ws[ak + 1][nB + in * 16 + nl];
      }
      for (int im = 0; im < 4; im++)
        for (int in = 0; in < 2; in++)
          acc[im][in] = wmma4(a[im], bb[in], acc[im][in]);
    }
  }

  const float* xb = x + (size_t)b * CCH * HWN + tt * 64;
  float* yb = y + (size_t)b * CCH * HWN + tt * 64;
  float lsum = 0.0f, lsq = 0.0f;
  for (int im = 0; im < 4; im++) {
    for (int in = 0; in < 2; in++) {
      int n = nB + in * 16 + nl;
      float bias = b2[n];
      for (int r = 0; r < 8; r++) {
        int m = im * 16 + r + 8 * half;
        float v = acc[im][in][r] + bias + xb[(size_t)n * HWN + m];
        yb[(size_t)n * HWN + m] = v;
        lsum += v; lsq += v * v;
      }
    }
  }
  atomicAdd(&stats[b * 2], lsum);
  atomicAdd(&stats[b * 2 + 1], lsq);
}

// ---------------------------------------------------------------------------
// Kernel 4: per-batch LayerNorm over (C,H,W) with elementwise affine, in place
// ---------------------------------------------------------------------------
__global__ void ln_kernel(float* __restrict__ y, const float* __restrict__ stats,
                          const float* __restrict__ lw, const float* __restrict__ lb) {
  const float invN = 1.0f / (float)(CCH * HWN);
  size_t total = (size_t)BATCH * CCH * HWN;
  size_t stride = (size_t)gridDim.x * blockDim.x;
  for (size_t idx = (size_t)blockIdx.x * blockDim.x + threadIdx.x; idx < total;
       idx += stride) {
    int b = (int)(idx >> 18);
    int chw = (int)(idx & ((1u << 18) - 1u));
    float mu = stats[2 * b] * invN;
    float var = stats[2 * b + 1] * invN - mu * mu;
    float r = rsqrtf(var + LN_EPS);
    y[idx] = (y[idx] - mu) * r * lw[chw] + lb[chw];
  }
}

// ---------------------------------------------------------------------------
extern "C" void kernel_launch(void* const* d_in, const int* in_sizes, int n_in,
                              void* d_out, int out_size, void* d_ws, size_t ws_size,
                              hipStream_t stream) {
  const float* x  = (const float*)d_in[0];
  const float* Wq = (const float*)d_in[1];
  const float* bq = (const float*)d_in[2];
  const float* Wk = (const float*)d_in[3];
  const float* bk = (const float*)d_in[4];
  const float* Wv = (const float*)d_in[5];
  const float* bv = (const float*)d_in[6];
  const float* W1 = (const float*)d_in[7];
  const float* b1 = (const float*)d_in[8];
  const float* W2 = (const float*)d_in[9];
  const float* b2 = (const float*)d_in[10];
  const float* lw = (const float*)d_in[11];
  const float* lb = (const float*)d_in[12];
  float* out = (float*)d_out;

  float* qkv   = (float*)d_ws;                        // 64*320*1024 floats
  float* attn  = qkv + (size_t)BATCH * MQKV * HWN;    // 64*256*1024 floats
  float* stats = attn + (size_t)BATCH * CCH * HWN;    // 128 floats

  zero_kernel<<<1, 128, 0, stream>>>(stats, 128);
  qkv_kernel<<<dim3(4, 5, BATCH), 256, 0, stream>>>(x, Wq, bq, Wk, bk, Wv, bv, qkv);
  attn_kernel<<<dim3(16, BATCH), 256, 0, stream>>>(qkv, attn);
  mlp_kernel<<<dim3(16, BATCH), 256, 0, stream>>>(attn, x, W1, b1, W2, b2, out, stats);
  ln_kernel<<<dim3(2048), 256, 0, stream>>>(out, stats, lw, lb);
}